// NodeModel_39865886441900
// MI455X (gfx1250) — compile-verified
//
#include <hip/hip_runtime.h>

// ---------------------------------------------------------------------------
// GNN NodeModel for MI455X (gfx1250), fp32 end-to-end using V_WMMA_F32_16X16X4_F32.
// See analysis: BN folded into affine (a,c) pairs so the edge->node scatter can
// move pre-BN sums; big matmuls run on the fp32 WMMA path; scatter uses
// global_atomic_add_f32 into an L2-resident [N,128] buffer.
// ---------------------------------------------------------------------------

constexpr int Nn = 100000;
constexpr int Ee = 1600000;
constexpr float EPSf = 1e-5f;

typedef float v2f __attribute__((ext_vector_type(2)));
typedef float v8f __attribute__((ext_vector_type(8)));

// ---- workspace layout (floats) --------------------------------------------
constexpr long OFF_SUMS = 0;                       // [N*128] scatter sums, reused as h3
constexpr long OFF_CNT  = OFF_SUMS + (long)Nn*128; // [N] degree
constexpr long OFF_BN1S = OFF_CNT + Nn;
constexpr long OFF_BN1Q = OFF_BN1S + 128;
constexpr long OFF_BN2S = OFF_BN1Q + 128;
constexpr long OFF_BN2Q = OFF_BN2S + 128;
constexpr long OFF_BN3S = OFF_BN2Q + 128;
constexpr long OFF_BN3Q = OFF_BN3S + 128;
constexpr long OFF_BN4S = OFF_BN3Q + 128;
constexpr long OFF_BN4Q = OFF_BN4S + 4;
constexpr long OFF_A1   = OFF_BN4Q + 4;            // memset covers [0, OFF_A1)
constexpr long OFF_C1   = OFF_A1 + 128;
constexpr long OFF_A2   = OFF_C1 + 128;
constexpr long OFF_C2   = OFF_A2 + 128;
constexpr long OFF_A3   = OFF_C2 + 128;
constexpr long OFF_C3   = OFF_A3 + 128;

// ---------------------------------------------------------------------------
// Pass 1: BN1 statistics over relu(W1@feat+b1) for all edges, + degree count.
// Thread t handles channel (t&127) for half the edges of each 128-edge chunk.
// ---------------------------------------------------------------------------
__global__ __launch_bounds__(256) void k_bn1_stats(
    const float* __restrict__ x, const int* __restrict__ ei,
    const float* __restrict__ ea,
    const float* __restrict__ w1, const float* __restrict__ b1,
    float* __restrict__ bn1s, float* __restrict__ bn1q, float* __restrict__ cnt)
{
  __shared__ float lf[128 * 4];
  const int t = threadIdx.x;
  const int c = t & 127;
  const int half = t >> 7;
  const float w0 = w1[c*3+0], wA = w1[c*3+1], wB = w1[c*3+2], bb = b1[c];
  float sum = 0.f, sq = 0.f;
  for (int base = blockIdx.x * 128; base < Ee; base += gridDim.x * 128) {
    __syncthreads();
    if (t < 128) {
      const int e = base + t;
      const int s = ei[Ee + e];               // src = edge_index[1][e]
      lf[t*4+0] = x[2*s];
      lf[t*4+1] = x[2*s+1];
      lf[t*4+2] = ea[e];
      atomicAdd(&cnt[s], 1.0f);               // degree for scatter_mean
    }
    __syncthreads();
    #pragma unroll 4
    for (int j = 0; j < 64; ++j) {
      const int el = half*64 + j;
      float h = fmaf(w0, lf[el*4+0], fmaf(wA, lf[el*4+1], fmaf(wB, lf[el*4+2], bb)));
      h = fmaxf(h, 0.f);
      sum += h; sq += h*h;
    }
  }
  atomicAdd(&bn1s[c], sum);
  atomicAdd(&bn1q[c], sq);
}

// ---------------------------------------------------------------------------
// BN finalize: a = g*rsqrt(var+eps), c = beta - mu*a   (biased variance)
// ---------------------------------------------------------------------------
__global__ void k_bn_finalize(const float* __restrict__ s, const float* __restrict__ q,
                              const float* __restrict__ g, const float* __restrict__ be,
                              float* __restrict__ a, float* __restrict__ c,
                              int nch, float dinv)
{
  const int i = blockIdx.x * blockDim.x + threadIdx.x;
  if (i < nch) {
    const float mu  = s[i] * dinv;
    const float var = q[i] * dinv - mu * mu;
    const float aa  = g[i] * rsqrtf(var + EPSf);
    a[i] = aa;
    c[i] = be[i] - mu * aa;
  }
}

// ---------------------------------------------------------------------------
// Edge layer-2 (the 52 GFLOP matmul): per 256-thread block, tile = 128 edges.
//   phase1: recompute h1, apply BN1 affine -> LDS A-tile [128][128]
//   phase2: 8 waves x (8 N-tiles x 32 k-steps) of v_wmma_f32_16x16x4_f32
//   phase3: bias+relu, BN2 stats in registers, atomic scatter into sums
// W2^T staged once per block in LDS in (k-pair, n) layout so each B fragment
// is one ds_load_b64.
// ---------------------------------------------------------------------------
__global__ __launch_bounds__(256) void k_edge_l2(
    const float* __restrict__ x, const int* __restrict__ ei,
    const float* __restrict__ ea,
    const float* __restrict__ w1, const float* __restrict__ b1,
    const float* __restrict__ a1, const float* __restrict__ c1,
    const float* __restrict__ w2, const float* __restrict__ b2,
    float* __restrict__ sums, float* __restrict__ bn2s, float* __restrict__ bn2q)
{
  extern __shared__ __attribute__((aligned(16))) float smem[];
  float* lds_a   = smem;               // [128][128] h1bn tile (row-major)
  float* lds_b   = smem + 16384;       // W2^T pairs: [(k>>1)][n][2]
  float* lds_w1  = lds_b + 16384;      // [128][4] = w1 row (3) + b1
  float* lds_aff = lds_w1 + 512;       // a1[128], c1[128]
  float* lds_b2  = lds_aff + 256;      // b2[128]
  int*   lds_src = (int*)(lds_b2 + 128); // [128]

  const int t = threadIdx.x;
  for (int i = t; i < 128*128; i += 256) {
    const int n = i >> 7, k = i & 127;
    lds_b[((k >> 1)*128 + n)*2 + (k & 1)] = w2[n*128 + k];   // B[k][n] = W2[n][k]
  }
  if (t < 128) {
    lds_w1[t*4+0] = w1[t*3+0]; lds_w1[t*4+1] = w1[t*3+1];
    lds_w1[t*4+2] = w1[t*3+2]; lds_w1[t*4+3] = b1[t];
    lds_aff[t] = a1[t]; lds_aff[128+t] = c1[t];
    lds_b2[t] = b2[t];
  }

  const int lane = t & 31, w = t >> 5;
  const int hi = lane >> 4, l15 = lane & 15;
  const int el = t >> 1, chalf = (t & 1) * 64;

  float ssum[8] = {0,0,0,0,0,0,0,0};
  float ssq[8]  = {0,0,0,0,0,0,0,0};

  for (int tile = blockIdx.x; tile < Ee/128; tile += gridDim.x) {
    const int e0 = tile * 128;
    __syncthreads();                       // protect LDS vs previous iteration
    // ---- phase 1: h1bn tile ----
    {
      const int e = e0 + el;
      const int s = ei[Ee + e];
      const float f0 = x[2*s], f1 = x[2*s+1], f2 = ea[e];
      if ((t & 1) == 0) lds_src[el] = s;
      // prefetch next tile's edge data into L2 (global_prefetch_b8)
      if (t < 32) __builtin_prefetch(&ea[e0 + gridDim.x*128 + t*4], 0, 1);
      #pragma unroll 4
      for (int c = chalf; c < chalf + 64; ++c) {
        float h = fmaf(lds_w1[c*4+0], f0,
                  fmaf(lds_w1[c*4+1], f1,
                  fmaf(lds_w1[c*4+2], f2, lds_w1[c*4+3])));
        h = fmaxf(h, 0.f);
        lds_a[el*128 + c] = fmaf(lds_aff[c], h, lds_aff[128 + c]);
      }
    }
    __syncthreads();
    // ---- phase 2: WMMA ----
    v8f acc[8];
    const v8f zero = {};
    #pragma unroll
    for (int n = 0; n < 8; ++n) acc[n] = zero;
    const float* arow = lds_a + (w*16 + l15)*128 + hi*2;
    for (int kk = 0; kk < 32; ++kk) {
      const v2f a = *(const v2f*)(arow + kk*4);
      const float* bbase = lds_b + ((kk*2 + hi)*128 + l15)*2;
      #pragma unroll
      for (int n = 0; n < 8; ++n) {
        const v2f b = *(const v2f*)(bbase + n*32);
        acc[n] = __builtin_amdgcn_wmma_f32_16x16x4_f32(
            false, a, false, b, (short)0, acc[n], false, false);
      }
    }
    // ---- phase 3: bias + relu, BN2 stats, scatter ----
    #pragma unroll
    for (int n = 0; n < 8; ++n) {
      const int ch = n*16 + l15;
      const float bb = lds_b2[ch];
      #pragma unroll
      for (int r = 0; r < 8; ++r) {
        const int rloc = w*16 + r + hi*8;    // C/D layout: M = r + 8*hi
        const float v = fmaxf(acc[n][r] + bb, 0.f);
        ssum[n] += v; ssq[n] += v*v;
        atomicAdd(&sums[(long)lds_src[rloc]*128 + ch], v);
      }
    }
  }
  #pragma unroll
  for (int n = 0; n < 8; ++n) {
    const int ch = n*16 + l15;
    atomicAdd(&bn2s[ch], ssum[n]);
    atomicAdd(&bn2q[ch], ssq[n]);
  }
}

// ---------------------------------------------------------------------------
// Node layer-1 WMMA: agg=(a2*sums+c2*cnt)/max(cnt,1) -> LDS; W3[:, :2]@x + b3
// initializes the accumulators (so K=128 exactly); h3 overwrites sums in-place
// (tiles are disjoint across blocks; reads of a tile complete before writes).
// ---------------------------------------------------------------------------
__global__ __launch_bounds__(256) void k_node_l1(
    const float* __restrict__ x, const float* __restrict__ cnt,
    const float* __restrict__ a2, const float* __restrict__ c2,
    const float* __restrict__ w3, const float* __restrict__ b3,
    float* __restrict__ sums,     // in: scatter sums; out: h3
    float* __restrict__ bn3s, float* __restrict__ bn3q)
{
  extern __shared__ __attribute__((aligned(16))) float smem[];
  float* lds_a   = smem;               // [128][128] normalized agg
  float* lds_b   = smem + 16384;       // W3[:,2:130]^T pairs
  float* lds_x   = lds_b + 16384;      // [128][2]
  float* lds_wh  = lds_x + 256;        // [128][4] = w3[:,0], w3[:,1], b3
  float* lds_aff = lds_wh + 512;       // a2[128], c2[128]

  const int t = threadIdx.x;
  for (int i = t; i < 128*128; i += 256) {
    const int n = i >> 7, k = i & 127;
    lds_b[((k >> 1)*128 + n)*2 + (k & 1)] = w3[n*130 + 2 + k];
  }
  if (t < 128) {
    lds_wh[t*4+0] = w3[t*130+0];
    lds_wh[t*4+1] = w3[t*130+1];
    lds_wh[t*4+2] = b3[t];
    lds_aff[t] = a2[t]; lds_aff[128+t] = c2[t];
  }

  const int lane = t & 31, w = t >> 5;
  const int hi = lane >> 4, l15 = lane & 15;
  const int el = t >> 1, chalf = (t & 1) * 64;

  float ssum[8] = {0,0,0,0,0,0,0,0};
  float ssq[8]  = {0,0,0,0,0,0,0,0};

  const int ntiles = (Nn + 127) >> 7;    // 782 (last tile has 32 rows)
  for (int tile = blockIdx.x; tile < ntiles; tile += gridDim.x) {
    const int n0 = tile * 128;
    __syncthreads();
    // ---- phase 1: normalized aggregate tile ----
    {
      const int node = n0 + el;
      float cv = 0.f, inv = 0.f;
      const bool ok = node < Nn;
      if (ok) { cv = cnt[node]; inv = 1.0f / fmaxf(cv, 1.0f); }
      if ((t & 1) == 0) {
        lds_x[el*2+0] = ok ? x[2*node]   : 0.f;
        lds_x[el*2+1] = ok ? x[2*node+1] : 0.f;
      }
      #pragma unroll 4
      for (int c = chalf; c < chalf + 64; ++c) {
        float v = 0.f;
        if (ok) {
          const float s = sums[(long)node*128 + c];
          v = fmaf(lds_aff[c], s, lds_aff[128 + c] * cv) * inv;
        }
        lds_a[el*128 + c] = v;
      }
    }
    __syncthreads();
    // ---- phase 2: init with x-columns, then WMMA over K=128 ----
    v8f acc[8];
    #pragma unroll
    for (int n = 0; n < 8; ++n) {
      const int ch = n*16 + l15;
      const float wh0 = lds_wh[ch*4+0], wh1 = lds_wh[ch*4+1], bb = lds_wh[ch*4+2];
      #pragma unroll
      for (int r = 0; r < 8; ++r) {
        const int rloc = w*16 + r + hi*8;
        acc[n][r] = fmaf(wh0, lds_x[rloc*2+0], fmaf(wh1, lds_x[rloc*2+1], bb));
      }
    }
    const float* arow = lds_a + (w*16 + l15)*128 + hi*2;
    for (int kk = 0; kk < 32; ++kk) {
      const v2f a = *(const v2f*)(arow + kk*4);
      const float* bbase = lds_b + ((kk*2 + hi)*128 + l15)*2;
      #pragma unroll
      for (int n = 0; n < 8; ++n) {
        const v2f b = *(const v2f*)(bbase + n*32);
        acc[n] = __builtin_amdgcn_wmma_f32_16x16x4_f32(
            false, a, false, b, (short)0, acc[n], false, false);
      }
    }
    // ---- phase 3: relu, BN3 stats, write h3 over sums ----
    #pragma unroll
    for (int n = 0; n < 8; ++n) {
      const int ch = n*16 + l15;
      #pragma unroll
      for (int r = 0; r < 8; ++r) {
        const int rloc = w*16 + r + hi*8;
        const int node = n0 + rloc;
        if (node < Nn) {
          const float v = fmaxf(acc[n][r], 0.f);
          ssum[n] += v; ssq[n] += v*v;
          sums[(long)node*128 + ch] = v;
        }
      }
    }
  }
  #pragma unroll
  for (int n = 0; n < 8; ++n) {
    const int ch = n*16 + l15;
    atomicAdd(&bn3s[ch], ssum[n]);
    atomicAdd(&bn3q[ch], ssq[n]);
  }
}

// ---------------------------------------------------------------------------
// Node layer-2 (128 -> 2) + BN4 statistics. Tiny FLOPs; one thread per node.
// ---------------------------------------------------------------------------
__global__ __launch_bounds__(256) void k_node_l2(
    const float* __restrict__ h3,
    const float* __restrict__ a3, const float* __restrict__ c3,
    const float* __restrict__ w4, const float* __restrict__ b4,
    float* __restrict__ out, float* __restrict__ bn4s, float* __restrict__ bn4q)
{
  __shared__ float lw[520];   // [0..255]=w4, [256..257]=b4, [258..385]=a3, [386..513]=c3
  const int t = threadIdx.x;
  if (t < 256) { lw[t] = w4[t]; lw[258 + t] = (t < 128) ? a3[t] : c3[t - 128]; }
  if (t < 2)   { lw[256 + t] = b4[t]; }
  __syncthreads();
  float s0 = 0.f, q0 = 0.f, s1 = 0.f, q1 = 0.f;
  for (int node = blockIdx.x*256 + t; node < Nn; node += gridDim.x*256) {
    const float* hr = h3 + (long)node*128;
    float v0 = lw[256], v1 = lw[257];
    #pragma unroll 4
    for (int c = 0; c < 128; ++c) {
      const float hn = fmaf(lw[258 + c], hr[c], lw[386 + c]);
      v0 = fmaf(hn, lw[c], v0);
      v1 = fmaf(hn, lw[128 + c], v1);
    }
    v0 = fmaxf(v0, 0.f); v1 = fmaxf(v1, 0.f);
    out[node*2+0] = v0; out[node*2+1] = v1;
    s0 += v0; q0 += v0*v0; s1 += v1; q1 += v1*v1;
  }
  atomicAdd(&bn4s[0], s0); atomicAdd(&bn4q[0], q0);
  atomicAdd(&bn4s[1], s1); atomicAdd(&bn4q[1], q1);
}

// ---------------------------------------------------------------------------
// Apply BN4 affine in place on the output.
// ---------------------------------------------------------------------------
__global__ void k_bn4_apply(float* __restrict__ out,
                            const float* __restrict__ bn4s, const float* __restrict__ bn4q,
                            const float* __restrict__ g, const float* __restrict__ be)
{
  const int i = blockIdx.x * blockDim.x + threadIdx.x;
  if (i < Nn) {
    const float dinv = 1.0f / (float)Nn;
    #pragma unroll
    for (int o = 0; o < 2; ++o) {
      const float mu  = bn4s[o] * dinv;
      const float var = bn4q[o] * dinv - mu*mu;
      const float a   = g[o] * rsqrtf(var + EPSf);
      const float c   = be[o] - mu*a;
      out[i*2+o] = fmaf(a, out[i*2+o], c);
    }
  }
}

// ---------------------------------------------------------------------------
extern "C" void kernel_launch(void* const* d_in, const int* in_sizes, int n_in,
                              void* d_out, int out_size, void* d_ws, size_t ws_size,
                              hipStream_t stream)
{
  (void)in_sizes; (void)n_in; (void)out_size; (void)ws_size;
  const float* x   = (const float*)d_in[0];
  const int*   ei  = (const int*)d_in[1];      // [2,E] int32; row1 = src
  const float* ea  = (const float*)d_in[2];
  // d_in[3] = u, d_in[4] = batch : unused by the reference computation
  const float* mw1  = (const float*)d_in[5];
  const float* mb1  = (const float*)d_in[6];
  const float* mg1  = (const float*)d_in[7];
  const float* mbe1 = (const float*)d_in[8];
  const float* mw2  = (const float*)d_in[9];
  const float* mb2  = (const float*)d_in[10];
  const float* mg2  = (const float*)d_in[11];
  const float* mbe2 = (const float*)d_in[12];
  const float* nw1  = (const float*)d_in[13];
  const float* nb1  = (const float*)d_in[14];
  const float* ng1  = (const float*)d_in[15];
  const float* nbe1 = (const float*)d_in[16];
  const float* nw2  = (const float*)d_in[17];
  const float* nb2  = (const float*)d_in[18];
  const float* ng2  = (const float*)d_in[19];
  const float* nbe2 = (const float*)d_in[20];

  float* ws  = (float*)d_ws;
  float* out = (float*)d_out;
  float* sums = ws + OFF_SUMS;
  float* cnt  = ws + OFF_CNT;
  float* bn1s = ws + OFF_BN1S; float* bn1q = ws + OFF_BN1Q;
  float* bn2s = ws + OFF_BN2S; float* bn2q = ws + OFF_BN2Q;
  float* bn3s = ws + OFF_BN3S; float* bn3q = ws + OFF_BN3Q;
  float* bn4s = ws + OFF_BN4S; float* bn4q = ws + OFF_BN4Q;
  float* a1v = ws + OFF_A1; float* c1v = ws + OFF_C1;
  float* a2v = ws + OFF_A2; float* c2v = ws + OFF_C2;
  float* a3v = ws + OFF_A3; float* c3v = ws + OFF_C3;

  // zero: sums, cnt, all BN stat accumulators (single contiguous region)
  hipMemsetAsync(d_ws, 0, (size_t)OFF_A1 * sizeof(float), stream);

  k_bn1_stats<<<1024, 256, 0, stream>>>(x, ei, ea, mw1, mb1, bn1s, bn1q, cnt);
  k_bn_finalize<<<1, 128, 0, stream>>>(bn1s, bn1q, mg1, mbe1, a1v, c1v, 128, 1.0f/(float)Ee);

  const size_t shB = (size_t)(16384 + 16384 + 512 + 256 + 128 + 128) * 4;  // 132 KB
  k_edge_l2<<<1280, 256, shB, stream>>>(x, ei, ea, mw1, mb1, a1v, c1v, mw2, mb2,
                                        sums, bn2s, bn2q);
  k_bn_finalize<<<1, 128, 0, stream>>>(bn2s, bn2q, mg2, mbe2, a2v, c2v, 128, 1.0f/(float)Ee);

  const size_t shD = (size_t)(16384 + 16384 + 256 + 512 + 256) * 4;        // 132 KB
  k_node_l1<<<782, 256, shD, stream>>>(x, cnt, a2v, c2v, nw1, nb1, sums, bn3s, bn3q);
  k_bn_finalize<<<1, 128, 0, stream>>>(bn3s, bn3q, ng1, nbe1, a3v, c3v, 128, 1.0f/(float)Nn);

  k_node_l2<<<512, 256, 0, stream>>>(sums, a3v, c3v, nw2, nb2, out, bn4s, bn4q);
  k_bn4_apply<<<(Nn + 255)/256, 256, 0, stream>>>(out, bn4s, bn4q, ng2, nbe2);
}